// GRU_Amp_Bug_11879879543425
// MI455X (gfx1250) — compile-verified
//
#include <hip/hip_runtime.h>
#include <math.h>

// GRU, H=128, S=512, B=256. fp32 (matches reference numerics).
//   K1 : W_f = W_ih @ lin_W, b_f = W_ih @ lin_b + b_ih — fragment-order output.
//   K1b: permute W_hh into fragment order.
//   K2 : gx_all = X @ W_f^T + b_f  (131072 x 384, K=128), v_wmma f32, NT stores.
//   K3 : batch-parallel recurrence, 16 WGs; W_hh fragments register-resident,
//        gx double-buffered in LDS via GLOBAL_LOAD_ASYNC_TO_LDS_B128 so the
//        next step's 24KB gx slice streams in behind the 96-WMMA chain.

#define H 128
#define S 512
#define BATCH 256
#define G 384        // 3*H
#define HSTRIDE 132  // LDS row pad: 132 mod 64 = 4 -> conflict-free A reads

typedef float v2f __attribute__((ext_vector_type(2)));
typedef float v8f __attribute__((ext_vector_type(8)));
typedef int   v4i __attribute__((vector_size(16)));   // builtin's pointee type

#define GLOBAL_AS __attribute__((address_space(1)))
#define LDS_AS    __attribute__((address_space(3)))

#if __has_builtin(__builtin_amdgcn_global_load_async_to_lds_b128)
#define USE_ASYNC 1
#else
#define USE_ASYNC 0
#endif

__device__ __forceinline__ void wait_asynccnt0() {
#if __has_builtin(__builtin_amdgcn_s_wait_asynccnt)
  __builtin_amdgcn_s_wait_asynccnt(0);
#else
  asm volatile("s_wait_asynccnt 0" ::: "memory");
#endif
}

// Fragment-order index for an n-major matrix used as WMMA B:
// B[k][n], n = 16*t + ln, k = 4*kt + r; lane = (r&1)*16 + ln, comp = r>>1.
__device__ __forceinline__ int frag_idx(int n, int k) {
  const int t  = n >> 4, ln = n & 15;
  const int kt = k >> 2, r  = k & 3;
  const int lane = ((r & 1) << 4) | ln;
  const int comp = r >> 1;
  return (((t * 32 + kt) * 32) + lane) * 2 + comp;
}

// ---------------------------------------------------------------------------
// K1: weight fusion, fragment-order output.
// ---------------------------------------------------------------------------
__global__ void fuse_weights_kernel(const float* __restrict__ lin_W,
                                    const float* __restrict__ lin_b,
                                    const float* __restrict__ W_ih,
                                    const float* __restrict__ b_ih,
                                    float* __restrict__ Wf_frag,
                                    float* __restrict__ b_f) {
  const int g = blockIdx.x;
  const int k = threadIdx.x;
  float acc = 0.0f;
  for (int d = 0; d < H; ++d)
    acc += W_ih[g * H + d] * lin_W[d * H + k];
  Wf_frag[frag_idx(g, k)] = acc;
  if (k == 0) {
    float bb = b_ih[g];
    for (int d = 0; d < H; ++d)
      bb += W_ih[g * H + d] * lin_b[d];
    b_f[g] = bb;
  }
}

__global__ void swizzle_whh_kernel(const float* __restrict__ W_hh,
                                   float* __restrict__ Whh_frag) {
  const int g = blockIdx.x;
  const int k = threadIdx.x;
  Whh_frag[frag_idx(g, k)] = W_hh[g * H + k];
}

// ---------------------------------------------------------------------------
// K2: gx = X @ W_f^T + b_f. Grid (1024,3), 256 thr = 8 waves, wave -> 16x128.
// X (67MB) is L2-resident across the 3x N-split; gx stored non-temporal.
// ---------------------------------------------------------------------------
__global__ void gx_gemm_kernel(const float* __restrict__ X,
                               const float* __restrict__ Wf_frag,
                               const float* __restrict__ b_f,
                               float* __restrict__ gx) {
  const int wave = threadIdx.x >> 5;
  const int lane = threadIdx.x & 31;
  const int ln   = lane & 15;
  const int hi   = lane >> 4;
  const int m0   = blockIdx.x * 128 + wave * 16;
  const int n0t  = blockIdx.y * 8;

  v8f acc[8];
#pragma unroll
  for (int t = 0; t < 8; ++t) acc[t] = {};

  const int arow = m0 + ln;
  for (int kt = 0; kt < 32; ++kt) {
    const int ka = kt * 4 + hi * 2;
    const v2f a = *(const v2f*)(X + (size_t)arow * H + ka);
#pragma unroll
    for (int t = 0; t < 8; ++t) {
      const int tt = n0t + t;
      const v2f b = *(const v2f*)(Wf_frag + (((tt * 32 + kt) * 32) + lane) * 2);
      acc[t] = __builtin_amdgcn_wmma_f32_16x16x4_f32(
          false, a, false, b, (short)0, acc[t], false, false);
    }
  }

#pragma unroll
  for (int t = 0; t < 8; ++t) {
    const int col  = (n0t + t) * 16 + ln;
    const float bv = b_f[col];
#pragma unroll
    for (int i = 0; i < 8; ++i) {
      const int row = m0 + i + hi * 8;
      __builtin_nontemporal_store(acc[t][i] + bv, gx + (size_t)row * G + col);
    }
  }
}

// ---------------------------------------------------------------------------
// K3: recurrence. 16 WGs x 256 thr. Wave j owns H columns [16j,16j+16).
// 96 B fragments in registers; h tile + gx double buffer in LDS.
// ---------------------------------------------------------------------------
__global__ void gru_recurrent_kernel(const float* __restrict__ gx,
                                     const float* __restrict__ h0,
                                     const float* __restrict__ Whh_frag,
                                     const float* __restrict__ b_hh,
                                     float* __restrict__ out) {
  __shared__ float hl[16 * HSTRIDE];
#if USE_ASYNC
  __shared__ float gxbuf[2][16 * G];   // 2 x 24KB double buffer
#endif

  const int tid  = threadIdx.x;
  const int wave = tid >> 5;
  const int lane = tid & 31;
  const int ln   = lane & 15;
  const int hi   = lane >> 4;
  const int bb   = blockIdx.x * 16;

#if USE_ASYNC
  // Issue one step's 16x384 contiguous gx slice: 6 x b128 per thread.
  auto issue_gx = [&](int s, int buf) {
    const float* src = gx + (size_t)s * BATCH * G + (size_t)bb * G;
#pragma unroll
    for (int c = 0; c < 6; ++c) {
      const int off = (c * 256 + tid) * 4;
      __builtin_amdgcn_global_load_async_to_lds_b128(
          (GLOBAL_AS v4i*)(src + off),
          (LDS_AS v4i*)(&gxbuf[buf][off]), 0, 0);
    }
  };
  issue_gx(0, 0);   // in flight while we stage h0 and preload fragments
#endif

  // Stage h0 into LDS (padded rows).
  for (int i = tid; i < 16 * H; i += 256) {
    const int m = i >> 7, k = i & 127;
    hl[m * HSTRIDE + k] = h0[(size_t)(bb + m) * H + k];
  }

  const int j   = wave;
  const int col = j * 16 + ln;
  const float bhr = b_hh[col];
  const float bhz = b_hh[H + col];
  const float bhn = b_hh[2 * H + col];

  // Preload all 96 B fragments into registers (loop-invariant).
  v2f bfr[32], bfz[32], bfn[32];
#pragma unroll
  for (int kt = 0; kt < 32; ++kt) {
    bfr[kt] = *(const v2f*)(Whh_frag + ((((0 * 8 + j) * 32 + kt) * 32) + lane) * 2);
    bfz[kt] = *(const v2f*)(Whh_frag + ((((1 * 8 + j) * 32 + kt) * 32) + lane) * 2);
    bfn[kt] = *(const v2f*)(Whh_frag + ((((2 * 8 + j) * 32 + kt) * 32) + lane) * 2);
  }
  __syncthreads();

  for (int s = 0; s < S; ++s) {
    // 6 independent accumulator chains (even/odd ktile per gate) for ILP.
    v8f cr0 = {}, cr1 = {}, cz0 = {}, cz1 = {}, cn0 = {}, cn1 = {};
#pragma unroll
    for (int kt = 0; kt < 32; kt += 2) {
      const int ka0 = kt * 4 + hi * 2;
      const v2f a0 = *(const v2f*)(hl + ln * HSTRIDE + ka0);
      const v2f a1 = *(const v2f*)(hl + ln * HSTRIDE + ka0 + 4);
      cr0 = __builtin_amdgcn_wmma_f32_16x16x4_f32(false, a0, false, bfr[kt],     (short)0, cr0, false, false);
      cz0 = __builtin_amdgcn_wmma_f32_16x16x4_f32(false, a0, false, bfz[kt],     (short)0, cz0, false, false);
      cn0 = __builtin_amdgcn_wmma_f32_16x16x4_f32(false, a0, false, bfn[kt],     (short)0, cn0, false, false);
      cr1 = __builtin_amdgcn_wmma_f32_16x16x4_f32(false, a1, false, bfr[kt + 1], (short)0, cr1, false, false);
      cz1 = __builtin_amdgcn_wmma_f32_16x16x4_f32(false, a1, false, bfz[kt + 1], (short)0, cz1, false, false);
      cn1 = __builtin_amdgcn_wmma_f32_16x16x4_f32(false, a1, false, bfn[kt + 1], (short)0, cn1, false, false);
    }
    const v8f cr = cr0 + cr1, cz = cz0 + cz1, cn = cn0 + cn1;

#if USE_ASYNC
    wait_asynccnt0();   // this wave's slice loads done; barrier makes all visible
#endif
    __syncthreads();    // also: all waves finished reading the h tile

#if USE_ASYNC
    const float* gxs = gxbuf[s & 1];
#else
    const float* gxs = gx + (size_t)s * BATCH * G + (size_t)bb * G;
#endif
#pragma unroll
    for (int i = 0; i < 8; ++i) {
      const int m = i + hi * 8;
      const float xr = gxs[m * G + col];
      const float xz = gxs[m * G + H + col];
      const float xn = gxs[m * G + 2 * H + col];
      const float hprev = hl[m * HSTRIDE + col];
      const float r = 1.0f / (1.0f + __expf(-(xr + cr[i] + bhr)));
      const float z = 1.0f / (1.0f + __expf(-(xz + cz[i] + bhz)));
      const float n = tanhf(xn + r * (cn[i] + bhn));
      const float hnew = (1.0f - z) * n + z * hprev;
      hl[m * HSTRIDE + col] = hnew;                  // wave-disjoint columns
      __builtin_nontemporal_store(hnew, out + ((size_t)s * BATCH + bb + m) * H + col);
    }

#if USE_ASYNC
    if (s + 1 < S) issue_gx(s + 1, (s + 1) & 1);     // prefetch next step
#else
    if (s + 1 < S)
      __builtin_prefetch(gx + (size_t)(s + 1) * BATCH * G + (size_t)bb * G + col, 0, 1);
#endif
    __syncthreads();    // h tile fully updated before next step's A reads
  }
}

// ---------------------------------------------------------------------------
extern "C" void kernel_launch(void* const* d_in, const int* in_sizes, int n_in,
                              void* d_out, int out_size, void* d_ws, size_t ws_size,
                              hipStream_t stream) {
  const float* X     = (const float*)d_in[0];
  const float* h     = (const float*)d_in[1];
  const float* lin_W = (const float*)d_in[2];
  const float* lin_b = (const float*)d_in[3];
  const float* W_ih  = (const float*)d_in[4];
  const float* W_hh  = (const float*)d_in[5];
  const float* b_ih  = (const float*)d_in[6];
  const float* b_hh  = (const float*)d_in[7];
  float* out = (float*)d_out;

  float* ws       = (float*)d_ws;
  float* Wf_frag  = ws;                       // 49152 floats
  float* b_f      = ws + G * H;               // 384
  float* Whh_frag = ws + G * H + 1024;        // 49152
  float* gx       = ws + 2 * G * H + 2048;    // 50,331,648 floats (~201MB)

  fuse_weights_kernel<<<G, H, 0, stream>>>(lin_W, lin_b, W_ih, b_ih, Wf_frag, b_f);
  swizzle_whh_kernel<<<G, H, 0, stream>>>(W_hh, Whh_frag);

  gx_gemm_kernel<<<dim3((S * BATCH) / 128, 3), 256, 0, stream>>>(X, Wf_frag, b_f, gx);

  gru_recurrent_kernel<<<BATCH / 16, 256, 0, stream>>>(gx, h, Whh_frag, b_hh, out);
}